// SinkLayer_39444979646958
// MI455X (gfx1250) — compile-verified
//
#include <hip/hip_runtime.h>

// Sinkhorn-style normalization collapsed to diagonal scalings:
//   out = diag(a) * (x+eps) * diag(b),  a = 1/((x+eps) b),  b = 1/((x+eps)^T a)
// 20 L2-resident f32 WMMA matvec passes + 1 HBM-bound elementwise pass.

typedef __attribute__((ext_vector_type(2))) float v2f;
typedef __attribute__((ext_vector_type(8))) float v8f;

#define SINK_EPS 0.001f
#define SINK_N   1024
#define SINK_B   32
#define SINK_IT  10

// ---- row pass: a[i] = 1 / sum_j (x[i,j]+eps) * b[j] -----------------------
// One wave computes 16 rows via V_WMMA_F32_16X16X4_F32 with B broadcast
// across the N dimension. Block = 256 threads = 8 waves = 128 rows.
// Grid = 32 batches * 8 = 256 blocks.
template <bool USE_B>
__global__ __launch_bounds__(256) void sink_rowpass(const float* __restrict__ x,
                                                    const float* __restrict__ bvec,
                                                    float* __restrict__ avec) {
  const int wave  = threadIdx.x >> 5;
  const int lane  = threadIdx.x & 31;
  const int m     = lane & 15;       // row within 16-row tile
  const int kg    = lane >> 4;       // K-group: 0 -> K{0,1}, 1 -> K{2,3}
  const int batch = blockIdx.x >> 3;
  const int rowb  = ((blockIdx.x & 7) << 7) + (wave << 4);

  const float* __restrict__ rp = x + ((size_t)batch * SINK_N + rowb + m) * SINK_N;
  const float* __restrict__ bp = bvec + batch * SINK_N;

  v8f c = {0.f, 0.f, 0.f, 0.f, 0.f, 0.f, 0.f, 0.f};
#pragma unroll 4
  for (int k = 0; k < SINK_N; k += 4) {
    const int kk = k + (kg << 1);
    v2f A, B;
    A.x = rp[kk]     + SINK_EPS;   // A[m, 2*kg]
    A.y = rp[kk + 1] + SINK_EPS;   // A[m, 2*kg+1]
    if (USE_B) { B.x = bp[kk]; B.y = bp[kk + 1]; }  // B[k, n] = b[k]  (all n)
    else       { B.x = 1.f;    B.y = 1.f; }
    c = __builtin_amdgcn_wmma_f32_16x16x4_f32(false, A, false, B,
                                              (short)0, c, false, false);
  }
  // D[:, n] identical for all n; N=0 lives in lane 0 (M=0..7) / lane 16 (M=8..15)
  if (m == 0) {
    float* ap = avec + batch * SINK_N + rowb + (kg << 3);
#pragma unroll
    for (int r = 0; r < 8; ++r) ap[r] = 1.0f / c[r];
  }
}

// ---- col pass: b[j] = 1 / sum_i a[i] * (x[i,j]+eps) -----------------------
// One wave computes 16 columns: A = broadcast(a) over M, B = 4x16 tile of x
// (rows striped across lanes -> 64B-contiguous loads per 16 lanes).
__global__ __launch_bounds__(256) void sink_colpass(const float* __restrict__ x,
                                                    const float* __restrict__ avec,
                                                    float* __restrict__ bvec) {
  const int wave  = threadIdx.x >> 5;
  const int lane  = threadIdx.x & 31;
  const int n     = lane & 15;       // column within 16-col tile
  const int kg    = lane >> 4;
  const int batch = blockIdx.x >> 3;
  const int jb    = (((blockIdx.x & 7) << 3) + wave) << 4;

  const float* __restrict__ cp = x + (size_t)batch * SINK_N * SINK_N + jb + n;
  const float* __restrict__ ap = avec + batch * SINK_N;

  v8f c = {0.f, 0.f, 0.f, 0.f, 0.f, 0.f, 0.f, 0.f};
#pragma unroll 2
  for (int i = 0; i < SINK_N; i += 4) {
    const int ii = i + (kg << 1);
    v2f A, B;
    A.x = ap[ii];                              // A[m, k] = a[k]  (all m)
    A.y = ap[ii + 1];
    B.x = cp[(size_t)ii * SINK_N]       + SINK_EPS;  // B[2*kg,   n] = x[ii,   j]
    B.y = cp[(size_t)(ii + 1) * SINK_N] + SINK_EPS;  // B[2*kg+1, n] = x[ii+1, j]
    c = __builtin_amdgcn_wmma_f32_16x16x4_f32(false, A, false, B,
                                              (short)0, c, false, false);
  }
  // All M rows of D identical; c[0] on lanes 0..15 holds v[jb + n]
  if (kg == 0) bvec[batch * SINK_N + jb + n] = 1.0f / c[0];
}

// ---- finalize: out[i,j] = (x[i,j]+eps) * a[i] * b[j] ----------------------
__global__ __launch_bounds__(256) void sink_finalize(const float* __restrict__ x,
                                                     const float* __restrict__ avec,
                                                     const float* __restrict__ bvec,
                                                     float* __restrict__ out) {
  const size_t f = (size_t)blockIdx.x * blockDim.x + threadIdx.x;  // float4 idx
  const size_t e = f << 2;
  const int batch = (int)(e >> 20);
  const int i     = (int)((e >> 10) & (SINK_N - 1));
  const int j     = (int)(e & (SINK_N - 1));

  const float4 xv = ((const float4*)x)[f];
  const float4 bv = *(const float4*)(bvec + batch * SINK_N + j);
  const float  a  = avec[batch * SINK_N + i];

  float4 o;
  o.x = (xv.x + SINK_EPS) * a * bv.x;
  o.y = (xv.y + SINK_EPS) * a * bv.y;
  o.z = (xv.z + SINK_EPS) * a * bv.z;
  o.w = (xv.w + SINK_EPS) * a * bv.w;
  ((float4*)out)[f] = o;
}

extern "C" void kernel_launch(void* const* d_in, const int* in_sizes, int n_in,
                              void* d_out, int out_size, void* d_ws, size_t ws_size,
                              hipStream_t stream) {
  (void)in_sizes; (void)n_in; (void)out_size; (void)ws_size;
  const float* x = (const float*)d_in[0];
  float* out = (float*)d_out;
  float* a = (float*)d_ws;                 // 32*1024 floats
  float* b = a + SINK_B * SINK_N;          // 32*1024 floats (256 KB total)

  const dim3 blk(256);
  const dim3 grd(SINK_B * 8);              // 256 blocks per matvec pass

  // iter 0 row pass uses implicit b = 1 (no init kernel needed)
  sink_rowpass<false><<<grd, blk, 0, stream>>>(x, b, a);
  sink_colpass<<<grd, blk, 0, stream>>>(x, a, b);
  for (int it = 1; it < SINK_IT; ++it) {
    sink_rowpass<true><<<grd, blk, 0, stream>>>(x, b, a);
    sink_colpass<<<grd, blk, 0, stream>>>(x, a, b);
  }

  const int nf4 = SINK_B * SINK_N * SINK_N / 4;       // 2^23 float4s
  sink_finalize<<<nf4 / 256, blk, 0, stream>>>(x, a, b, out);
}